// InvariantPointAttention_50809463112125
// MI455X (gfx1250) — compile-verified
//
#include <hip/hip_runtime.h>

// Shapes: B=1, L=512, C_S=384, H=12, D=32
#define L512 512
#define CS 384
#define NH 12
#define HD 32
#define LL (512 * 512) // elements per (h) plane of bias/gate

typedef float v2f __attribute__((ext_vector_type(2)));
typedef float v8f __attribute__((ext_vector_type(8)));

// D(16x16,f32) = A(16x4,f32) * B(4x16,f32) + C   [V_WMMA_F32_16X16X4_F32]
__device__ __forceinline__ v8f wmma4(v2f a, v2f b, v8f c) {
  return __builtin_amdgcn_wmma_f32_16x16x4_f32(
      /*neg_a=*/false, a, /*neg_b=*/false, b,
      /*c_mod=*/(short)0, c, /*reuse_a=*/false, /*reuse_b=*/false);
}

// CDNA5 async LDS<->global DMA (ASYNCcnt-tracked), per ISA 15.18.3.
__device__ __forceinline__ void async_load_b128(unsigned lds_off,
                                                const void* gaddr) {
  asm volatile("global_load_async_to_lds_b128 %0, %1, off"
               :: "v"(lds_off), "v"(gaddr) : "memory");
}
__device__ __forceinline__ void async_store_b128(void* gaddr,
                                                 unsigned lds_off) {
  asm volatile("global_store_async_from_lds_b128 %0, %1, off"
               :: "v"(gaddr), "v"(lds_off) : "memory");
}
__device__ __forceinline__ unsigned lds_off32(const void* p) {
  return (unsigned)(uintptr_t)p;  // low 32 bits of generic addr = LDS offset
}

// ---------------------------------------------------------------------------
// K1: q,k,v = s @ [Wq | Wkv] + bias.  One wave -> 16(M) x 64(N) tile.
// ---------------------------------------------------------------------------
__global__ __launch_bounds__(256) void qkv_kernel(
    const float* __restrict__ s, const float* __restrict__ Wq,
    const float* __restrict__ bq, const float* __restrict__ Wkv,
    const float* __restrict__ bkv, float* __restrict__ qw,
    float* __restrict__ kw, float* __restrict__ vw) {
  const int wave = threadIdx.x >> 5, lane = threadIdx.x & 31;
  const int idx = blockIdx.x * 8 + wave;        // 0..575
  const int mTile = idx / 18, nStrip = idx % 18;
  const int mn = lane & 15, kb = (lane >> 4) << 1, m8 = (lane >> 4) << 3;
  const int row = mTile * 16 + mn;
  const int colBase = nStrip * 64;

  const float* W; const float* bvec; float* outp; int ldw, wcol0, ocol0;
  if (colBase < 384) {
    W = Wq;  bvec = bq;  ldw = 384; wcol0 = colBase;       outp = qw; ocol0 = colBase;
  } else if (colBase < 768) {
    W = Wkv; bvec = bkv; ldw = 768; wcol0 = colBase - 384; outp = kw; ocol0 = colBase - 384;
  } else {
    W = Wkv; bvec = bkv; ldw = 768; wcol0 = colBase - 384; outp = vw; ocol0 = colBase - 768;
  }

  v8f acc[4] = {};
  for (int k0 = 0; k0 < CS; k0 += 4) {
    v2f a = *(const v2f*)&s[row * CS + k0 + kb];
#pragma unroll
    for (int t = 0; t < 4; ++t) {
      const int wc = wcol0 + t * 16 + mn;
      v2f b;
      b.x = W[(k0 + kb) * ldw + wc];
      b.y = W[(k0 + kb + 1) * ldw + wc];
      acc[t] = wmma4(a, b, acc[t]);
    }
  }
#pragma unroll
  for (int t = 0; t < 4; ++t) {
    const float bv = bvec[wcol0 + t * 16 + mn];
    const int oc = ocol0 + t * 16 + mn;
#pragma unroll
    for (int r = 0; r < 8; ++r)
      outp[(mTile * 16 + r + m8) * CS + oc] = acc[t][r] + bv;
  }
}

// ---------------------------------------------------------------------------
// K2: fused bias/gate projection over z (403 MB stream -> roofline limiter).
// Each wave owns 16 z-rows; double-buffered global_load_async_to_lds_b128
// staging (no workgroup barriers in the k-loop, only s_wait_asynccnt).
// Results transposed in LDS and written with global_store_async_from_lds.
// ---------------------------------------------------------------------------
#define ZROWS 128
#define ZSTR 36    // floats per LDS row: 144B, 16B-aligned, conflict padding
#define OSTR 132   // plane stride for out staging: 528B, 16B-aligned
__global__ __launch_bounds__(256) void biasgate_kernel(
    const float* __restrict__ z, const float* __restrict__ Wb,
    const float* __restrict__ bb, const float* __restrict__ Wg,
    const float* __restrict__ bg, float* __restrict__ biasO,
    float* __restrict__ gateO) {
  __shared__ float zt[2][ZROWS * ZSTR];  // 2 x 18432 B
  __shared__ float outt[24 * OSTR];      // 12672 B
  const int tid = threadIdx.x, wave = tid >> 5, lane = tid & 31;
  const size_t rbase = (size_t)blockIdx.x * ZROWS;
  const int mn = lane & 15, kb = (lane >> 4) << 1, m8 = (lane >> 4) << 3;

  // Prologue: async-stage chunk 0 of this wave's 16 rows (4 b128 per lane).
#pragma unroll
  for (int i = 0; i < 4; ++i) {
    const int u = lane + 32 * i;            // 0..127 16B units
    const int r = u >> 3, c = (u & 7) * 4;  // row 0..15, col floats
    async_load_b128(lds_off32(&zt[0][(wave * 16 + r) * ZSTR + c]),
                    &z[(rbase + wave * 16 + r) * CS + c]);
  }

  v8f accB = {}, accG = {};
  for (int k0 = 0; k0 < CS; k0 += 32) {
    const int buf = (k0 >> 5) & 1;
    if (k0 + 32 < CS) {
      // issue next chunk into the other buffer, then wait for current chunk
#pragma unroll
      for (int i = 0; i < 4; ++i) {
        const int u = lane + 32 * i;
        const int r = u >> 3, c = (u & 7) * 4;
        async_load_b128(lds_off32(&zt[buf ^ 1][(wave * 16 + r) * ZSTR + c]),
                        &z[(rbase + wave * 16 + r) * CS + k0 + 32 + c]);
      }
      asm volatile("s_wait_asynccnt 0x4" ::: "memory");
    } else {
      asm volatile("s_wait_asynccnt 0x0" ::: "memory");
    }
#pragma unroll
    for (int kc = 0; kc < 8; ++kc) {
      const int kk = kc * 4 + kb;
      v2f a = *(const v2f*)&zt[buf][(wave * 16 + mn) * ZSTR + kk];
      const int kg = k0 + kk;
      v2f b1, b2;
      if (mn < NH) {
        b1.x = Wb[kg * NH + mn]; b1.y = Wb[(kg + 1) * NH + mn];
        b2.x = Wg[kg * NH + mn]; b2.y = Wg[(kg + 1) * NH + mn];
      } else {
        b1.x = b1.y = b2.x = b2.y = 0.0f;
      }
      accB = wmma4(a, b1, accB);
      accG = wmma4(a, b2, accG);
    }
  }

  // Transpose into planar staging (plane-major, coalesced), then async store.
  __syncthreads();
  if (mn < NH) {
    const float bbv = bb[mn], bgv = bg[mn];
#pragma unroll
    for (int r = 0; r < 8; ++r) {
      const int mrow = wave * 16 + r + m8;  // 0..127
      outt[mn * OSTR + mrow] = accB[r] + bbv;
      const float gx = accG[r] + bgv;
      outt[(NH + mn) * OSTR + mrow] = 1.0f / (1.0f + __expf(-gx));
    }
  }
  __syncthreads();
  // 24 planes x 128 floats = 768 b128 units; 3 per thread, LDS -> global DMA.
#pragma unroll
  for (int i = 0; i < 3; ++i) {
    const int u = tid + 256 * i;            // 0..767
    const int plane = u >> 5, mu = (u & 31) * 4;
    float* g = (plane < NH)
                   ? &biasO[(size_t)plane * LL + rbase + mu]
                   : &gateO[(size_t)(plane - NH) * LL + rbase + mu];
    async_store_b128(g, lds_off32(&outt[plane * OSTR + mu]));
  }
  // S_ENDPGM performs an implicit wait-idle (drains ASYNCcnt).
}

// ---------------------------------------------------------------------------
// K3: per (head, 16 query rows): logits=QK^T*scale+bias, mask, softmax,
// *gate, then attn @ V.  Logits tile 16x512 lives in padded LDS.
// ---------------------------------------------------------------------------
#define LTS 516
__global__ __launch_bounds__(256) void attn_kernel(
    const float* __restrict__ qm, const float* __restrict__ km,
    const float* __restrict__ vm, const float* __restrict__ biasI,
    const float* __restrict__ gateI, const int* __restrict__ mask,
    float* __restrict__ o) {
  __shared__ float lt[16 * LTS];  // 33024 B
  __shared__ float lo[16 * 33];   // o accumulator, padded
  const int tid = threadIdx.x, wave = tid >> 5, lane = tid & 31;
  const int h = blockIdx.y, lbase = blockIdx.x * 16;
  const int mn = lane & 15, kb = (lane >> 4) << 1, m8 = (lane >> 4) << 3;
  const float scale = 0.17677669529663687f;  // 32^-0.5

  // Phase 1: logits (each wave: column tiles ct = wave, wave+8, ...)
  for (int ct = wave; ct < 32; ct += 8) {
    v8f acc = {};
#pragma unroll
    for (int d0 = 0; d0 < HD; d0 += 4) {
      v2f a = *(const v2f*)&qm[(size_t)(lbase + mn) * CS + h * HD + d0 + kb];
      v2f b = *(const v2f*)&km[(size_t)(ct * 16 + mn) * CS + h * HD + d0 + kb];
      acc = wmma4(a, b, acc);
    }
    const int j = ct * 16 + mn;
    const int mj = mask[j];
#pragma unroll
    for (int r = 0; r < 8; ++r) {
      const int m = r + m8, i = lbase + m;
      float lg;
      if (mj == 0 || mask[i] == 0)
        lg = -1.0e9f;
      else
        lg = acc[r] * scale + biasI[(size_t)h * LL + (size_t)i * L512 + j];
      lt[m * LTS + j] = lg;
    }
  }
  __syncthreads();

  // Phase 2: softmax per row (wave handles rows wave, wave+8), fold in gate
#pragma unroll
  for (int rr = 0; rr < 2; ++rr) {
    const int r = wave + rr * 8, i = lbase + r;
    float mx = -3.0e38f;
    for (int j = lane; j < L512; j += 32) mx = fmaxf(mx, lt[r * LTS + j]);
#pragma unroll
    for (int off = 16; off > 0; off >>= 1) mx = fmaxf(mx, __shfl_xor(mx, off, 32));
    float sm = 0.0f;
    for (int j = lane; j < L512; j += 32) {
      const float e = __expf(lt[r * LTS + j] - mx);
      lt[r * LTS + j] = e;
      sm += e;
    }
#pragma unroll
    for (int off = 16; off > 0; off >>= 1) sm += __shfl_xor(sm, off, 32);
    const float inv = 1.0f / sm;
    for (int j = lane; j < L512; j += 32)
      lt[r * LTS + j] *= inv * gateI[(size_t)h * LL + (size_t)i * L512 + j];
  }
  __syncthreads();
  for (int idx = tid; idx < 16 * 33; idx += 256) lo[idx] = 0.0f;
  __syncthreads();

  // Phase 3: o = attn @ V; 8 waves = {2 n-tiles} x {4 K-slices of 128}
  {
    const int nt = wave & 1, ks = wave >> 1;
    v8f acc = {};
    for (int k0 = ks * 128; k0 < ks * 128 + 128; k0 += 4) {
      v2f a = *(const v2f*)&lt[mn * LTS + k0 + kb];
      v2f b;
      b.x = vm[(size_t)(k0 + kb) * CS + h * HD + nt * 16 + mn];
      b.y = vm[(size_t)(k0 + kb + 1) * CS + h * HD + nt * 16 + mn];
      acc = wmma4(a, b, acc);
    }
    const int d = nt * 16 + mn;
#pragma unroll
    for (int r = 0; r < 8; ++r) atomicAdd(&lo[(r + m8) * 33 + d], acc[r]);
  }
  __syncthreads();
  for (int idx = tid; idx < 16 * HD; idx += 256) {
    const int m = idx >> 5, d = idx & 31;
    o[(size_t)(lbase + m) * CS + h * HD + d] = lo[m * 33 + d];
  }
}

// ---------------------------------------------------------------------------
// K4: out = o @ Wout + bout.  Wave -> 16x64 tile.
// ---------------------------------------------------------------------------
__global__ __launch_bounds__(256) void outproj_kernel(
    const float* __restrict__ o, const float* __restrict__ Wout,
    const float* __restrict__ bout, float* __restrict__ out) {
  const int wave = threadIdx.x >> 5, lane = threadIdx.x & 31;
  const int idx = blockIdx.x * 8 + wave;  // 0..191
  const int mTile = idx / 6, nStrip = idx % 6;
  const int mn = lane & 15, kb = (lane >> 4) << 1, m8 = (lane >> 4) << 3;
  const int row = mTile * 16 + mn;

  v8f acc[4] = {};
  for (int k0 = 0; k0 < CS; k0 += 4) {
    v2f a = *(const v2f*)&o[row * CS + k0 + kb];
#pragma unroll
    for (int t = 0; t < 4; ++t) {
      const int c = nStrip * 64 + t * 16 + mn;
      v2f b;
      b.x = Wout[(k0 + kb) * CS + c];
      b.y = Wout[(k0 + kb + 1) * CS + c];
      acc[t] = wmma4(a, b, acc[t]);
    }
  }
#pragma unroll
  for (int t = 0; t < 4; ++t) {
    const int c = nStrip * 64 + t * 16 + mn;
    const float bv = bout[c];
#pragma unroll
    for (int r = 0; r < 8; ++r)
      out[(mTile * 16 + r + m8) * CS + c] = acc[t][r] + bv;
  }
}

extern "C" void kernel_launch(void* const* d_in, const int* in_sizes, int n_in,
                              void* d_out, int out_size, void* d_ws,
                              size_t ws_size, hipStream_t stream) {
  const float* s    = (const float*)d_in[0];
  const float* z    = (const float*)d_in[1];
  const int*   mask = (const int*)d_in[2];
  const float* Wq   = (const float*)d_in[3];
  const float* bq   = (const float*)d_in[4];
  const float* Wkv  = (const float*)d_in[5];
  const float* bkv  = (const float*)d_in[6];
  const float* Wb   = (const float*)d_in[7];
  const float* bb   = (const float*)d_in[8];
  const float* Wg   = (const float*)d_in[9];
  const float* bg   = (const float*)d_in[10];
  const float* Wout = (const float*)d_in[11];
  const float* bout = (const float*)d_in[12];
  float* out = (float*)d_out;

  float* ws = (float*)d_ws;
  float* qw    = ws;                        // 512*384
  float* kw    = ws + 196608;               // 512*384
  float* vw    = ws + 393216;               // 512*384
  float* biasw = ws + 589824;               // 12*512*512
  float* gatew = ws + 589824 + 3145728;     // 12*512*512
  float* ow    = ws + 589824 + 2 * 3145728; // 512*384

  qkv_kernel<<<72, 256, 0, stream>>>(s, Wq, bq, Wkv, bkv, qw, kw, vw);
  biasgate_kernel<<<2048, 256, 0, stream>>>(z, Wb, bb, Wg, bg, biasw, gatew);
  attn_kernel<<<dim3(32, 12), 256, 0, stream>>>(qw, kw, vw, biasw, gatew, mask, ow);
  outproj_kernel<<<24, 256, 0, stream>>>(ow, Wout, bout, out);
}